// WideBlock_19756849562051
// MI455X (gfx1250) — compile-verified
//
#include <hip/hip_runtime.h>

#define NSIDE 128
#define NPIX  (12 * NSIDE * NSIDE)   // 196608
#define BATCH 4
#define FINC  2
#define N2    (NPIX / 4)             // 49152

typedef __attribute__((ext_vector_type(2))) float v2f;
typedef __attribute__((ext_vector_type(8))) float v8f;

// ---------------------------------------------------------------------------
// Chebyshev step on the fine graph:  Tout = alpha*lap(Tprev) - (FIRST?0:Tm2)
// lap(y)[b,n,f] = sum_j w[n,j] * y[b, idx[n,j], f]   (Fin = 2, float2 per pixel)
// All gathers stay inside the 192MB L2 (each T slab is 6.3 MB).
// ---------------------------------------------------------------------------
template<int NN, bool FIRST>
__global__ void __launch_bounds__(256)
cheb_step_kernel(const float* __restrict__ Tprev,
                 const float* __restrict__ Tm2,
                 const int*   __restrict__ idx,
                 const float* __restrict__ w,
                 float*       __restrict__ Tout)
{
    int t = blockIdx.x * blockDim.x + threadIdx.x;
    if (t >= BATCH * NPIX) return;
    int n = t % NPIX, b = t / NPIX;
    const float2* Tp = (const float2*)Tprev;
    const int*   ip = idx + (size_t)n * NN;
    const float* wp = w   + (size_t)n * NN;
    float ax = 0.f, ay = 0.f;
    #pragma unroll
    for (int j = 0; j < NN; ++j) {
        int   k  = ip[j];
        float wt = wp[j];
        float2 v = Tp[(size_t)b * NPIX + k];
        ax = fmaf(wt, v.x, ax);
        ay = fmaf(wt, v.y, ay);
    }
    float2 o;
    if (FIRST) { o.x = ax; o.y = ay; }
    else {
        float2 m2 = ((const float2*)Tm2)[(size_t)b * NPIX + n];
        o.x = 2.f * ax - m2.x;
        o.y = 2.f * ay - m2.y;
    }
    ((float2*)Tout)[(size_t)b * NPIX + n] = o;
}

// ---------------------------------------------------------------------------
// Stage-A combine: out = pool4( LN( relu( sum_k T_k @ W_k + bias ) ) )
// One wave32 per coarse pixel; lane = output feature (F=32) or 2 feats (F=64).
// T_k values are wave-uniform (scalar loads); LN via wave shfl_xor reduction.
// ---------------------------------------------------------------------------
template<int K, int F>
__global__ void __launch_bounds__(128)
combineA_kernel(const float* __restrict__ maps,   // T0: [B,NPIX,2]
                const float* __restrict__ Tbuf,   // T1.. slots, stride B*NPIX*2
                const float* __restrict__ W,      // [K,2,F]
                const float* __restrict__ bias,
                const float* __restrict__ gamma,
                const float* __restrict__ beta,
                float*       __restrict__ hout)   // [B,N2,F]
{
    const int lane = threadIdx.x & 31;
    const int wid  = threadIdx.x >> 5;
    const int gid  = blockIdx.x * 4 + wid;          // over B*N2
    if (gid >= BATCH * N2) return;
    const int nc = gid % N2, b = gid / N2;
    constexpr int NF = F / 32;                       // features per lane
    constexpr size_t SLOT = (size_t)BATCH * NPIX * FINC;

    float pooled[NF];
    #pragma unroll
    for (int q = 0; q < NF; ++q) pooled[q] = 0.f;

    for (int c = 0; c < 4; ++c) {
        const int n = nc * 4 + c;
        float acc[NF];
        #pragma unroll
        for (int q = 0; q < NF; ++q) acc[q] = bias[lane + 32 * q];

        const float2 t0 = ((const float2*)maps)[(size_t)b * NPIX + n];
        #pragma unroll
        for (int q = 0; q < NF; ++q) {
            const int o = lane + 32 * q;
            acc[q] = fmaf(t0.x, W[0 * F + o], acc[q]);
            acc[q] = fmaf(t0.y, W[1 * F + o], acc[q]);
        }
        #pragma unroll
        for (int k = 1; k < K; ++k) {
            const float2 tk =
                ((const float2*)(Tbuf + (size_t)(k - 1) * SLOT))[(size_t)b * NPIX + n];
            #pragma unroll
            for (int q = 0; q < NF; ++q) {
                const int o = lane + 32 * q;
                acc[q] = fmaf(tk.x, W[(k * FINC + 0) * F + o], acc[q]);
                acc[q] = fmaf(tk.y, W[(k * FINC + 1) * F + o], acc[q]);
            }
        }
        // relu + LN statistics across the F features of this fine pixel
        float s = 0.f, s2 = 0.f;
        #pragma unroll
        for (int q = 0; q < NF; ++q) {
            acc[q] = fmaxf(acc[q], 0.f);
            s  += acc[q];
            s2 += acc[q] * acc[q];
        }
        #pragma unroll
        for (int m = 1; m < 32; m <<= 1) {
            s  += __shfl_xor(s,  m, 32);
            s2 += __shfl_xor(s2, m, 32);
        }
        const float mean = s / (float)F;
        const float var  = s2 / (float)F - mean * mean;
        const float rstd = rsqrtf(var + 1e-5f);
        #pragma unroll
        for (int q = 0; q < NF; ++q) {
            const int o = lane + 32 * q;
            pooled[q] += (acc[q] - mean) * rstd * gamma[o] + beta[o];
        }
    }
    #pragma unroll
    for (int q = 0; q < NF; ++q)
        hout[((size_t)b * N2 + nc) * F + lane + 32 * q] = 0.25f * pooled[q];
}

// ---------------------------------------------------------------------------
// Coarse-graph Laplacian gather:  g[b,n,:] = sum_j w8s[n,j] * h[b, idx8s[n,j], :]
// One thread per (b, n, 4-feature chunk); float4 vectorized, L2-resident.
// ---------------------------------------------------------------------------
template<int F>
__global__ void __launch_bounds__(256)
gatherS_kernel(const float* __restrict__ h,
               const int*   __restrict__ idx,
               const float* __restrict__ w,
               float*       __restrict__ g)
{
    constexpr int FQ = F / 4;
    int t = blockIdx.x * blockDim.x + threadIdx.x;
    if (t >= BATCH * N2 * FQ) return;
    int fq = t % FQ;
    int bn = t / FQ;
    int n  = bn % N2, b = bn / N2;
    const int*   ip = idx + (size_t)n * 8;
    const float* wp = w   + (size_t)n * 8;
    float4 acc = make_float4(0.f, 0.f, 0.f, 0.f);
    #pragma unroll
    for (int j = 0; j < 8; ++j) {
        int   k  = ip[j];
        float wt = wp[j];
        const float4 v = *(const float4*)(h + ((size_t)b * N2 + k) * F + fq * 4);
        acc.x = fmaf(wt, v.x, acc.x);
        acc.y = fmaf(wt, v.y, acc.y);
        acc.z = fmaf(wt, v.z, acc.z);
        acc.w = fmaf(wt, v.w, acc.w);
    }
    *(float4*)(g + (size_t)bn * F + fq * 4) = acc;
}

// ---------------------------------------------------------------------------
// Stage-B conv + ReLU + LN via V_WMMA_F32_16X16X4_F32 (full fp32 precision).
//   out[:, coff:coff+F] = LN( relu( h @ W0 + g @ W1 + bias ) )
// Each wave owns a 16-row block x all F columns (F/16 accumulator tiles).
// A operand (16x4 f32): lanes 0-15 rows, VGPR0/1 = K pair, lane>=16 -> K+2.
// B operand (4x16 f32): lane%16 = column, same K split.
// LN across features done in-register: row M lives in VGPR (M%8) of one
// 16-lane half, so xor masks 1,2,4,8 reduce one full row.
// ---------------------------------------------------------------------------
template<int F>
__global__ void __launch_bounds__(128)
conv2_wmma_kernel(const float* __restrict__ h,     // [M,F]
                  const float* __restrict__ g,     // [M,F]
                  const float* __restrict__ Wb,    // [2,F,F]
                  const float* __restrict__ bias,
                  const float* __restrict__ gamma,
                  const float* __restrict__ beta,
                  float*       __restrict__ out,   // [M,128]
                  int coff)
{
    constexpr int NC = F / 16;                     // column tiles
    constexpr int M  = BATCH * N2;                 // 196608, exact /16
    const int lane = threadIdx.x & 31;
    const int wid  = threadIdx.x >> 5;
    const int tile = blockIdx.x * 4 + wid;         // over M/16 (exact: EXEC stays all-1)
    if (tile >= M / 16) return;
    const int rowBase = tile * 16;
    const int l16 = lane & 15;
    const int kh  = (lane >> 4) * 2;               // 0 or 2

    v8f C[NC];
    #pragma unroll
    for (int c = 0; c < NC; ++c)
        #pragma unroll
        for (int v = 0; v < 8; ++v) C[c][v] = 0.f;

    const float* W0 = Wb;
    const float* W1 = Wb + F * F;
    const float* Ah = h + (size_t)(rowBase + l16) * F;
    const float* Ag = g + (size_t)(rowBase + l16) * F;

    #pragma unroll
    for (int src = 0; src < 2; ++src) {
        const float* A = src ? Ag : Ah;
        const float* W = src ? W1 : W0;
        #pragma unroll
        for (int k0 = 0; k0 < F; k0 += 4) {
            v2f a;
            a.x = A[k0 + kh + 0];
            a.y = A[k0 + kh + 1];
            #pragma unroll
            for (int c = 0; c < NC; ++c) {
                v2f bmat;
                bmat.x = W[(k0 + kh + 0) * F + c * 16 + l16];
                bmat.y = W[(k0 + kh + 1) * F + c * 16 + l16];
                C[c] = __builtin_amdgcn_wmma_f32_16x16x4_f32(
                    false, a, false, bmat, (short)0, C[c], false, false);
            }
        }
    }

    // bias + relu, then LayerNorm over the F features of each of the 8 rows
    float vals[NC][8];
    float s[8], s2[8];
    #pragma unroll
    for (int v = 0; v < 8; ++v) { s[v] = 0.f; s2[v] = 0.f; }
    #pragma unroll
    for (int c = 0; c < NC; ++c) {
        const float bcol = bias[c * 16 + l16];
        #pragma unroll
        for (int v = 0; v < 8; ++v) {
            float x = fmaxf(C[c][v] + bcol, 0.f);
            vals[c][v] = x;
            s[v]  += x;
            s2[v] += x * x;
        }
    }
    #pragma unroll
    for (int m = 1; m < 16; m <<= 1) {
        #pragma unroll
        for (int v = 0; v < 8; ++v) {
            s[v]  += __shfl_xor(s[v],  m, 32);
            s2[v] += __shfl_xor(s2[v], m, 32);
        }
    }
    #pragma unroll
    for (int v = 0; v < 8; ++v) {
        const float mean = s[v] / (float)F;
        const float var  = s2[v] / (float)F - mean * mean;
        const float rstd = rsqrtf(var + 1e-5f);
        const int   row  = rowBase + v + ((lane >> 4) << 3);
        #pragma unroll
        for (int c = 0; c < NC; ++c) {
            const int col = c * 16 + l16;
            out[(size_t)row * 128 + coff + col] =
                (vals[c][v] - mean) * rstd * gamma[col] + beta[col];
        }
    }
}

// ---------------------------------------------------------------------------
extern "C" void kernel_launch(void* const* d_in, const int* in_sizes, int n_in,
                              void* d_out, int out_size, void* d_ws, size_t ws_size,
                              hipStream_t stream)
{
    const float* maps  = (const float*)d_in[0];
    const float* w8    = (const float*)d_in[1];
    const float* w20   = (const float*)d_in[2];
    const float* w8s   = (const float*)d_in[3];
    const int*   idx8  = (const int*)d_in[4];
    const int*   idx20 = (const int*)d_in[5];
    const int*   idx8s = (const int*)d_in[6];
    const float* W1a = (const float*)d_in[7],  *b1a = (const float*)d_in[8];
    const float* g1a = (const float*)d_in[9],  *be1a = (const float*)d_in[10];
    const float* W1b = (const float*)d_in[11], *b1b = (const float*)d_in[12];
    const float* g1b = (const float*)d_in[13], *be1b = (const float*)d_in[14];
    const float* W2a = (const float*)d_in[15], *b2a = (const float*)d_in[16];
    const float* g2a = (const float*)d_in[17], *be2a = (const float*)d_in[18];
    const float* W2b = (const float*)d_in[19], *b2b = (const float*)d_in[20];
    const float* g2b = (const float*)d_in[21], *be2b = (const float*)d_in[22];
    const float* W3a = (const float*)d_in[23], *b3a = (const float*)d_in[24];
    const float* g3a = (const float*)d_in[25], *be3a = (const float*)d_in[26];
    const float* W3b = (const float*)d_in[27], *b3b = (const float*)d_in[28];
    const float* g3b = (const float*)d_in[29], *be3b = (const float*)d_in[30];
    float* out = (float*)d_out;
    float* ws  = (float*)d_ws;

    const size_t SLOT = (size_t)BATCH * NPIX * FINC;      // 1,572,864 floats
    float* T  = ws;                                       // 7 slots (T1..T7)
    float* H  = ws + 7 * SLOT;                            // B*N2*64 floats
    float* G  = H + (size_t)BATCH * N2 * 64;              // B*N2*64 floats
    float* h1 = H;                                        // [B,N2,32]
    float* h2 = H + (size_t)BATCH * N2 * 32;              // [B,N2,32]
    float* h3 = H;                                        // [B,N2,64] (reuse after 1&2)

    const int gridFine = (BATCH * NPIX + 255) / 256;      // cheb passes
    const int gridC    = (BATCH * N2 + 3) / 4;            // combineA (4 waves/blk)
    const int gridW    = (BATCH * N2 / 16 + 3) / 4;       // wmma tiles (4 waves/blk)
    const int gridG32  = (BATCH * N2 * 8  + 255) / 256;
    const int gridG64  = (BATCH * N2 * 16 + 255) / 256;

    // ---- Branches 1 & 2 share the idx8 Chebyshev sequence: T1..T5 ----
    cheb_step_kernel<8, true ><<<gridFine, 256, 0, stream>>>(maps,       nullptr,    idx8, w8, T + 0 * SLOT);
    cheb_step_kernel<8, false><<<gridFine, 256, 0, stream>>>(T + 0*SLOT, maps,       idx8, w8, T + 1 * SLOT);
    cheb_step_kernel<8, false><<<gridFine, 256, 0, stream>>>(T + 1*SLOT, T + 0*SLOT, idx8, w8, T + 2 * SLOT);
    cheb_step_kernel<8, false><<<gridFine, 256, 0, stream>>>(T + 2*SLOT, T + 1*SLOT, idx8, w8, T + 3 * SLOT);
    cheb_step_kernel<8, false><<<gridFine, 256, 0, stream>>>(T + 3*SLOT, T + 2*SLOT, idx8, w8, T + 4 * SLOT);

    combineA_kernel<4, 32><<<gridC, 128, 0, stream>>>(maps, T, W1a, b1a, g1a, be1a, h1);
    combineA_kernel<6, 32><<<gridC, 128, 0, stream>>>(maps, T, W2a, b2a, g2a, be2a, h2);

    gatherS_kernel<32><<<gridG32, 256, 0, stream>>>(h1, idx8s, w8s, G);
    conv2_wmma_kernel<32><<<gridW, 128, 0, stream>>>(h1, G, W1b, b1b, g1b, be1b, out, 0);
    gatherS_kernel<32><<<gridG32, 256, 0, stream>>>(h2, idx8s, w8s, G);
    conv2_wmma_kernel<32><<<gridW, 128, 0, stream>>>(h2, G, W2b, b2b, g2b, be2b, out, 32);

    // ---- Branch 3: idx20 Chebyshev sequence T1..T7 (reuses T region) ----
    cheb_step_kernel<20, true ><<<gridFine, 256, 0, stream>>>(maps,       nullptr,    idx20, w20, T + 0 * SLOT);
    cheb_step_kernel<20, false><<<gridFine, 256, 0, stream>>>(T + 0*SLOT, maps,       idx20, w20, T + 1 * SLOT);
    cheb_step_kernel<20, false><<<gridFine, 256, 0, stream>>>(T + 1*SLOT, T + 0*SLOT, idx20, w20, T + 2 * SLOT);
    cheb_step_kernel<20, false><<<gridFine, 256, 0, stream>>>(T + 2*SLOT, T + 1*SLOT, idx20, w20, T + 3 * SLOT);
    cheb_step_kernel<20, false><<<gridFine, 256, 0, stream>>>(T + 3*SLOT, T + 2*SLOT, idx20, w20, T + 4 * SLOT);
    cheb_step_kernel<20, false><<<gridFine, 256, 0, stream>>>(T + 4*SLOT, T + 3*SLOT, idx20, w20, T + 5 * SLOT);
    cheb_step_kernel<20, false><<<gridFine, 256, 0, stream>>>(T + 5*SLOT, T + 4*SLOT, idx20, w20, T + 6 * SLOT);

    combineA_kernel<8, 64><<<gridC, 128, 0, stream>>>(maps, T, W3a, b3a, g3a, be3a, h3);

    gatherS_kernel<64><<<gridG64, 256, 0, stream>>>(h3, idx8s, w8s, G);
    conv2_wmma_kernel<64><<<gridW, 128, 0, stream>>>(h3, G, W3b, b3b, g3b, be3b, out, 64);
}